// Encoder_8495445311558
// MI455X (gfx1250) — compile-verified
//
#include <hip/hip_runtime.h>
#include <stdint.h>

typedef __attribute__((ext_vector_type(2))) float v2f;
typedef __attribute__((ext_vector_type(8))) float v8f;

#define N_NODES 50000
#define N_EDGES 800000
#define NFEAT   256
#define NHID    128
#define LATENT  64
#define MT      5          // M-tiles per wave; 50000 = (16*5) * 625

// Guaranteed hardware f32 atomic add (no CAS fallback): non-returning
// global_atomic_add_f32, tracked with STOREcnt; S_ENDPGM's implicit
// wait-idle guarantees completion before the kernel retires.
__device__ __forceinline__ void gadd_f32(float* p, float v) {
    asm volatile("global_atomic_add_f32 %0, %1, off"
                 :: "v"((unsigned long long)(uintptr_t)p), "v"(v)
                 : "memory");
}

// ---------------------------------------------------------------------------
// GEMM1: h_lin = x @ W1   ([N_NODES,256] x [256,128] -> [N_NODES,128])
// One wave computes MT=5 16x16 C tiles (shared B fragment, 5 independent
// WMMA chains). 8 waves cover N=128; grid.x = N_NODES/(16*MT) = 625.
// ---------------------------------------------------------------------------
__global__ __launch_bounds__(256) void gemm1_wmma(const float* __restrict__ A,
                                                  const float* __restrict__ B,
                                                  float* __restrict__ C) {
    const int lane = threadIdx.x & 31;
    const int wave = threadIdx.x >> 5;
    const int m0   = blockIdx.x * (16 * MT);
    const int n0   = wave * 16;
    const int half = lane >> 4;          // lanes 16-31 handle K+2,K+3
    const int r    = lane & 15;          // A row / B,C column within tile
    const int kk   = half * 2;

    const float* a0 = A + (size_t)(m0 + r) * NFEAT + kk;
    const float* bp = B + (size_t)kk * NHID + n0 + r;

    v8f acc[MT];
    #pragma unroll
    for (int t = 0; t < MT; ++t) acc[t] = (v8f){};

    for (int k = 0; k < NFEAT; k += 4) {
        v2f b;                            // B rows k+kk, k+kk+1, col n0+r
        b.x = bp[0];
        b.y = bp[NHID];
        bp += 4 * NHID;
        #pragma unroll
        for (int t = 0; t < MT; ++t) {
            v2f a = *(const v2f*)(a0 + (size_t)t * 16 * NFEAT + k);
            acc[t] = __builtin_amdgcn_wmma_f32_16x16x4_f32(false, a, false, b,
                                                           (short)0, acc[t],
                                                           false, false);
        }
    }
    const int mb = m0 + half * 8;
    #pragma unroll
    for (int t = 0; t < MT; ++t) {
        #pragma unroll
        for (int i = 0; i < 8; ++i)
            C[(size_t)(mb + t * 16 + i) * NHID + n0 + r] = acc[t][i];
    }
}

// ---------------------------------------------------------------------------
// GEMM2: t = h @ [Wmu | Wlv]  ([N_NODES,128] x [128,128] -> [N_NODES,128])
// Columns 0..63 from Wmu, 64..127 from Wlv (per-wave uniform selection).
// ---------------------------------------------------------------------------
__global__ __launch_bounds__(256) void gemm2_wmma(const float* __restrict__ H,
                                                  const float* __restrict__ Wmu,
                                                  const float* __restrict__ Wlv,
                                                  float* __restrict__ C) {
    const int lane = threadIdx.x & 31;
    const int wave = threadIdx.x >> 5;
    const int m0   = blockIdx.x * (16 * MT);
    const int n0   = wave * 16;
    const int half = lane >> 4;
    const int r    = lane & 15;
    const int kk   = half * 2;

    const float* Bsel = (n0 < LATENT) ? Wmu : Wlv;     // uniform per wave
    const float* bp   = Bsel + (size_t)kk * LATENT + (n0 & (LATENT - 1)) + r;
    const float* a0   = H + (size_t)(m0 + r) * NHID + kk;

    v8f acc[MT];
    #pragma unroll
    for (int t = 0; t < MT; ++t) acc[t] = (v8f){};

    for (int k = 0; k < NHID; k += 4) {
        v2f b;
        b.x = bp[0];
        b.y = bp[LATENT];
        bp += 4 * LATENT;
        #pragma unroll
        for (int t = 0; t < MT; ++t) {
            v2f a = *(const v2f*)(a0 + (size_t)t * 16 * NHID + k);
            acc[t] = __builtin_amdgcn_wmma_f32_16x16x4_f32(false, a, false, b,
                                                           (short)0, acc[t],
                                                           false, false);
        }
    }
    const int mb = m0 + half * 8;
    #pragma unroll
    for (int t = 0; t < MT; ++t) {
        #pragma unroll
        for (int i = 0; i < 8; ++i)
            C[(size_t)(mb + t * 16 + i) * NHID + n0 + r] = acc[t][i];
    }
}

// ---------------------------------------------------------------------------
// Edge scatter (width 128): one wave32 per edge.
// Gather = one coalesced 512B row (float4/lane); scatter = 4 hardware
// global_atomic_add_f32 per lane into one contiguous 512B destination row.
// ---------------------------------------------------------------------------
__global__ __launch_bounds__(256) void edge_scatter128(const float* __restrict__ Hin,
                                                       const int* __restrict__ src,
                                                       const int* __restrict__ dst,
                                                       const float* __restrict__ w,
                                                       float* __restrict__ out) {
    const int e = blockIdx.x * (blockDim.x >> 5) + (threadIdx.x >> 5);
    if (e >= N_EDGES) return;
    const int   lane = threadIdx.x & 31;
    const int   s    = src[e];
    const int   d    = dst[e];
    const float ww   = w[e];

    const float4 v = *(const float4*)(Hin + (size_t)s * NHID + lane * 4);
    float* o = out + (size_t)d * NHID + lane * 4;
    gadd_f32(o + 0, v.x * ww);
    gadd_f32(o + 1, v.y * ww);
    gadd_f32(o + 2, v.z * ww);
    gadd_f32(o + 3, v.w * ww);
}

// h = relu(agg + b1), in place over N_NODES*128 elements
__global__ __launch_bounds__(256) void bias_relu128(float* __restrict__ h,
                                                    const float* __restrict__ b) {
    const size_t i = (size_t)blockIdx.x * blockDim.x + threadIdx.x;
    const float v = h[i] + b[i & (NHID - 1)];
    h[i] = v > 0.f ? v : 0.f;
}

// d_out[0 : N*64)       = agg2[:, 0:64]   + bmu
// d_out[N*64 : 2*N*64)  = agg2[:, 64:128] + blv
__global__ __launch_bounds__(256) void writeout(const float* __restrict__ agg2,
                                                const float* __restrict__ bmu,
                                                const float* __restrict__ blv,
                                                float* __restrict__ out) {
    const size_t idx = (size_t)blockIdx.x * blockDim.x + threadIdx.x; // N_NODES*64
    const size_t i = idx >> 6;
    const int    j = (int)(idx & 63);
    out[idx]                        = agg2[i * NHID + j] + bmu[j];
    out[(size_t)N_NODES * 64 + idx] = agg2[i * NHID + LATENT + j] + blv[j];
}

extern "C" void kernel_launch(void* const* d_in, const int* in_sizes, int n_in,
                              void* d_out, int out_size, void* d_ws, size_t ws_size,
                              hipStream_t stream) {
    (void)in_sizes; (void)n_in; (void)out_size; (void)ws_size;
    const float* x   = (const float*)d_in[0];
    const int*   src = (const int*)d_in[1];
    const int*   dst = (const int*)d_in[2];
    const float* ew  = (const float*)d_in[3];
    const float* W1  = (const float*)d_in[4];
    const float* b1  = (const float*)d_in[5];
    const float* Wmu = (const float*)d_in[6];
    const float* bmu = (const float*)d_in[7];
    const float* Wlv = (const float*)d_in[8];
    const float* blv = (const float*)d_in[9];
    float* out = (float*)d_out;

    const size_t nodeBufElems = (size_t)N_NODES * NHID;        // 6.4M floats
    float* bufA = (float*)d_ws;                                // h_lin, then t
    float* bufB = bufA + nodeBufElems;                         // agg1 -> h
    float* bufC = bufB + nodeBufElems;                         // agg2

    const dim3 blk(256);
    const unsigned gemmGrid = N_NODES / (16 * MT);             // 625

    // Layer 1: GEMM -> zero agg -> scatter -> bias+relu
    gemm1_wmma<<<dim3(gemmGrid), blk, 0, stream>>>(x, W1, bufA);
    hipMemsetAsync(bufB, 0, nodeBufElems * sizeof(float), stream);
    edge_scatter128<<<dim3(N_EDGES / 8), blk, 0, stream>>>(bufA, src, dst, ew, bufB);
    bias_relu128<<<dim3((unsigned)(nodeBufElems / 256)), blk, 0, stream>>>(bufB, b1);

    // Layer 2 (mu & logvar fused): GEMM -> zero agg -> scatter -> bias + write
    gemm2_wmma<<<dim3(gemmGrid), blk, 0, stream>>>(bufB, Wmu, Wlv, bufA);
    hipMemsetAsync(bufC, 0, nodeBufElems * sizeof(float), stream);
    edge_scatter128<<<dim3(N_EDGES / 8), blk, 0, stream>>>(bufA, src, dst, ew, bufC);
    writeout<<<dim3((unsigned)((size_t)N_NODES * 64 / 256)), blk, 0, stream>>>(bufC, bmu, blv, out);
}